// SiamFC__86260123174436
// MI455X (gfx1250) — compile-verified
//
#include <hip/hip_runtime.h>

// SiamFC cross-correlation via shift-sum GEMM on CDNA5 WMMA.
//
//   Q[m][s]   = sum_c z[n][c][m] * x[n][c][s]      (m = kh*6+kw in [0,36), s in [0,676))
//   out[oh,ow]= sum_{kh,kw} Q[kh*6+kw][(oh+kh)*26 + (ow+kw)]
//
// Per-sample GEMM: M=48 (36 padded), N=676 (43 tiles of 16), K=256.
// One workgroup (8 wave32) per sample; V_WMMA_F32_16X16X4_F32 for f32-exact math.

typedef __attribute__((ext_vector_type(2))) float v2f;
typedef __attribute__((ext_vector_type(8))) float v8f;

#define NSAMP   256
#define CHN     256
#define KPOS    36          // 6*6 kernel positions
#define MPAD    48          // 3 M-tiles of 16
#define NPIX    676         // 26*26 search pixels
#define WXX     26
#define OHW     21
#define NOUT    441         // 21*21
#define ZSTRIDE 260         // CHN + 4 pad -> conflict-free ds_load_b64 A-fragments
#define NTILES  43          // ceil(676/16)
#define NWAVES  8

__global__ __launch_bounds__(256)
void siamfc_xcorr_wmma(const float* __restrict__ z,
                       const float* __restrict__ x,
                       float* __restrict__ out) {
    __shared__ float zlds[MPAD * ZSTRIDE];  // zlds[kpos][c], rows 36..47 zeroed
    __shared__ float outLds[NOUT];

    const int n     = blockIdx.x;
    const int tid   = threadIdx.x;
    const int wave  = tid >> 5;
    const int lane  = tid & 31;
    const int nl    = lane & 15;     // tile column (pixel) / A-matrix row
    const int khalf = lane >> 4;     // 0 -> K{0,1}, 1 -> K{2,3} of the 4-wide K slab

    // ---- stage z transposed into LDS: zlds[kpos][c] (thread tid = channel) ----
    {
        const float* zsrc = z + (size_t)(n * CHN + tid) * KPOS; // 36 contiguous floats
        float zr[KPOS];
        #pragma unroll
        for (int k4 = 0; k4 < KPOS; k4 += 4) {
            float4 v = *(const float4*)(zsrc + k4);
            zr[k4 + 0] = v.x; zr[k4 + 1] = v.y; zr[k4 + 2] = v.z; zr[k4 + 3] = v.w;
        }
        #pragma unroll
        for (int k = 0; k < KPOS; ++k) zlds[k * ZSTRIDE + tid] = zr[k];
        #pragma unroll
        for (int k = KPOS; k < MPAD; ++k) zlds[k * ZSTRIDE + tid] = 0.0f; // zero M-pad rows
    }
    for (int p = tid; p < NOUT; p += 256) outLds[p] = 0.0f;
    __syncthreads();

    const float* xs = x + (size_t)n * CHN * NPIX;

    // ---- GEMM: each wave owns one N-tile (16 pixels), all 3 M-tiles ----
    for (int ntile = wave; ntile < NTILES; ntile += NWAVES) {
        const int n0    = ntile * 16;
        const int s_raw = n0 + nl;
        const int s     = (s_raw < NPIX) ? s_raw : (NPIX - 1);  // clamp pad lanes in-bounds

        v8f acc0 = {}; v8f acc1 = {}; v8f acc2 = {};

        // B-fragment base: this lane reads channels (c0 + 2*khalf) and +1 at pixel s
        const float* xb = xs + (size_t)(khalf * 2) * NPIX + s;

        #pragma unroll 8
        for (int c0 = 0; c0 < CHN; c0 += 4) {
            v2f b;
            b.x = xb[(size_t)c0 * NPIX];            // K = c0 + 2*khalf
            b.y = xb[(size_t)c0 * NPIX + NPIX];     // K = c0 + 2*khalf + 1
            const int ka = c0 + khalf * 2;
            v2f a0 = *(const v2f*)&zlds[(0  + nl) * ZSTRIDE + ka];  // M-tile 0, row nl
            v2f a1 = *(const v2f*)&zlds[(16 + nl) * ZSTRIDE + ka];  // M-tile 1
            v2f a2 = *(const v2f*)&zlds[(32 + nl) * ZSTRIDE + ka];  // M-tile 2
            acc0 = __builtin_amdgcn_wmma_f32_16x16x4_f32(false, a0, false, b, (short)0, acc0, false, false);
            acc1 = __builtin_amdgcn_wmma_f32_16x16x4_f32(false, a1, false, b, (short)0, acc1, false, false);
            acc2 = __builtin_amdgcn_wmma_f32_16x16x4_f32(false, a2, false, b, (short)0, acc2, false, false);
        }

        // ---- scatter Q -> out tile: Q[m][s] feeds at most one (oh,ow) ----
        if (s_raw < NPIX) {
            const int r  = s_raw / WXX;
            const int wc = s_raw % WXX;
            #pragma unroll
            for (int mt = 0; mt < 3; ++mt) {
                v8f a = (mt == 0) ? acc0 : (mt == 1) ? acc1 : acc2;
                #pragma unroll
                for (int v = 0; v < 8; ++v) {
                    const int m = mt * 16 + v + khalf * 8;   // C/D layout: lane half selects M+8
                    if (m < KPOS) {
                        const int kh = m / 6, kw = m % 6;
                        const int oh = r - kh, ow = wc - kw;
                        if (oh >= 0 && oh < OHW && ow >= 0 && ow < OHW)
                            atomicAdd(&outLds[oh * OHW + ow], a[v]);   // ds_add_f32
                    }
                }
            }
        }
    }

    __syncthreads();
    float* op = out + (size_t)n * NOUT;
    for (int p = tid; p < NOUT; p += 256) op[p] = outLds[p];
}

extern "C" void kernel_launch(void* const* d_in, const int* in_sizes, int n_in,
                              void* d_out, int out_size, void* d_ws, size_t ws_size,
                              hipStream_t stream) {
    (void)in_sizes; (void)n_in; (void)d_ws; (void)ws_size; (void)out_size;
    const float* z = (const float*)d_in[0];  // (256,256,6,6) f32
    const float* x = (const float*)d_in[1];  // (256,256,26,26) f32
    float* out = (float*)d_out;              // (256,1,21,21) f32
    siamfc_xcorr_wmma<<<dim3(NSAMP), dim3(256), 0, stream>>>(z, x, out);
}